// SecondaryCapsuleLayer_42949672961316
// MI455X (gfx1250) — compile-verified
//
#include <hip/hip_runtime.h>
#include <math.h>

// Problem constants (from reference setup_inputs)
#define C_CH   1024   // in_channels
#define N_CAP  32     // num_units
#define U_SZ   64     // unit_size
#define B_SZ   32     // batch
#define I_SZ   64     // in_units (K of the GEMM)
#define M_ROWS 2048   // N_CAP * U_SZ
#define TILE_F 65536  // M_ROWS * B_SZ  (floats per channel of u_hat)
#define NUM_IT 3

typedef float v2f __attribute__((ext_vector_type(2)));
typedef float v8f __attribute__((ext_vector_type(8)));

// ---------------------------------------------------------------------------
// Stage 1: u_hatT[c][m][b] = sum_i W[c][m][i] * x[b][i][c],  m = n*64+u
// One block per channel c. x[:, :, c] staged in LDS, 8 waves x 16 m-tiles.
// WMMA f32 16x16x4: A = W rows (16x4), B = x slice (4x16), D = 16x16 f32.
// ---------------------------------------------------------------------------
__global__ __launch_bounds__(256) void uhat_wmma_kernel(
    const float* __restrict__ x, const float* __restrict__ W,
    float* __restrict__ uhat) {
  __shared__ float Xs[B_SZ][I_SZ + 2];  // 66-float rows: bank-conflict padding
  const int c = blockIdx.x;
  const int t = threadIdx.x;

  // cooperative load of x[:, :, c] -> Xs[b][i]
  #pragma unroll
  for (int j = 0; j < 8; ++j) {
    int e = t + 256 * j;               // 0..2047
    int b = e >> 6, i = e & 63;
    Xs[b][i] = x[(size_t)(b * I_SZ + i) * C_CH + c];
  }
  __syncthreads();

  const int lane = t & 31;
  const int wave = t >> 5;
  const int half = lane >> 4;          // K-half selector (A/B layout)
  const int l16  = lane & 15;

  const float* Abase = W + (size_t)c * (M_ROWS * I_SZ);
  float*       Obase = uhat + (size_t)c * TILE_F;

  for (int j = 0; j < 16; ++j) {
    const int mt  = wave * 16 + j;     // m-tile 0..127
    const int row = mt * 16 + l16;     // A row for this lane
    const float* Ap = Abase + (size_t)row * I_SZ + half * 2;

    v8f acc0 = {0.f,0.f,0.f,0.f,0.f,0.f,0.f,0.f};
    v8f acc1 = {0.f,0.f,0.f,0.f,0.f,0.f,0.f,0.f};

    #pragma unroll
    for (int k0 = 0; k0 < I_SZ; k0 += 4) {
      // A lane data: W[c][row][k0 + half*2 + {0,1}]
      v2f a = *(const v2f*)(Ap + k0);
      const int kb = k0 + half * 2;
      // B lane data: B[k][n] = Xs[n][k]; n = l16 (+16 for second output tile)
      v2f b0 = *(const v2f*)&Xs[l16][kb];
      v2f b1 = *(const v2f*)&Xs[l16 + 16][kb];
      acc0 = __builtin_amdgcn_wmma_f32_16x16x4_f32(false, a, false, b0,
                                                   (short)0, acc0, false, false);
      acc1 = __builtin_amdgcn_wmma_f32_16x16x4_f32(false, a, false, b1,
                                                   (short)0, acc1, false, false);
    }

    // D layout: vgpr r, lanes0-15 -> M=r, lanes16-31 -> M=r+8, N=lane%16
    #pragma unroll
    for (int r = 0; r < 8; ++r) {
      const int orow = mt * 16 + r + half * 8;
      Obase[(size_t)orow * B_SZ + l16]      = acc0[r];
      Obase[(size_t)orow * B_SZ + l16 + 16] = acc1[r];
    }
  }
}

// ---------------------------------------------------------------------------
// Routing helper kernels (all HBM-bandwidth bound streaming; deterministic)
// ---------------------------------------------------------------------------
__global__ void zero_bij_kernel(float* __restrict__ bij) {
  int i = blockIdx.x * blockDim.x + threadIdx.x;
  if (i < C_CH * N_CAP) bij[i] = 0.0f;
}

__global__ void softmax_kernel(const float* __restrict__ bij,
                               float* __restrict__ cij) {
  int c = blockIdx.x * blockDim.x + threadIdx.x;
  if (c >= C_CH) return;
  float vals[N_CAP];
  float m = -INFINITY;
  #pragma unroll
  for (int n = 0; n < N_CAP; ++n) { vals[n] = bij[c * N_CAP + n]; m = fmaxf(m, vals[n]); }
  float s = 0.0f;
  #pragma unroll
  for (int n = 0; n < N_CAP; ++n) { vals[n] = expf(vals[n] - m); s += vals[n]; }
  float inv = 1.0f / s;
  #pragma unroll
  for (int n = 0; n < N_CAP; ++n) cij[c * N_CAP + n] = vals[n] * inv;
}

// partial[chunk][n][e] = sum_{c in chunk} cij[c][n] * uhatT[c][n*64.. ][e]
// grid: blockIdx = n*32 + chunk  (32 chunks of 32 channels)
__global__ __launch_bounds__(256) void s_partial_kernel(
    const float* __restrict__ uhat, const float* __restrict__ cij,
    float* __restrict__ partial) {
  const int n = blockIdx.x >> 5;
  const int chunk = blockIdx.x & 31;
  const int t = threadIdx.x;
  float acc[8] = {0.f,0.f,0.f,0.f,0.f,0.f,0.f,0.f};
  for (int cc = 0; cc < 32; ++cc) {
    const int c = chunk * 32 + cc;
    const float w = cij[c * N_CAP + n];
    const float* tile = uhat + (size_t)c * TILE_F + n * 2048;
    #pragma unroll
    for (int j = 0; j < 8; ++j) acc[j] = fmaf(w, tile[t + 256 * j], acc[j]);
  }
  float* po = partial + ((size_t)chunk * N_CAP + n) * 2048;
  #pragma unroll
  for (int j = 0; j < 8; ++j) po[t + 256 * j] = acc[j];
}

// sT[n*2048 + u*32 + b] = sum over 32 chunks
__global__ void s_reduce_kernel(const float* __restrict__ partial,
                                float* __restrict__ sT) {
  const int e = blockIdx.x * blockDim.x + threadIdx.x;  // 0..65535
  float s = 0.0f;
  #pragma unroll
  for (int ch = 0; ch < 32; ++ch) s += partial[(size_t)ch * (N_CAP * 2048) + e];
  sT[e] = s;
}

// squash over capsule axis n: mag_sq[b,u] = sum_n s[b,n,u]^2
// v = s * mag / (1 + mag_sq). Writes vT[n][u][b] (ws) and out[b][n][u][1].
__global__ void squash_kernel(const float* __restrict__ sT,
                              float* __restrict__ vT,
                              float* __restrict__ out) {
  const int t = blockIdx.x * blockDim.x + threadIdx.x;  // 0..2047 = u*32+b
  const int b = t & 31, u = t >> 5;
  float sv[N_CAP];
  float msq = 0.0f;
  #pragma unroll
  for (int n = 0; n < N_CAP; ++n) {
    float s = sT[n * 2048 + t];
    sv[n] = s;
    msq = fmaf(s, s, msq);
  }
  const float scale = sqrtf(msq) / (1.0f + msq);
  #pragma unroll
  for (int n = 0; n < N_CAP; ++n) {
    float v = sv[n] * scale;
    vT[n * 2048 + t] = v;
    out[(size_t)(b * N_CAP + n) * U_SZ + u] = v;
  }
}

// b_ij[c,n] += (1/B) * sum_{u,b} uhatT[c][n*64+u][b] * vT[n][u][b]
// grid: blockIdx = c*32 + n
__global__ __launch_bounds__(256) void agree_kernel(
    const float* __restrict__ uhat, const float* __restrict__ vT,
    float* __restrict__ bij) {
  const int c = blockIdx.x >> 5;
  const int n = blockIdx.x & 31;
  const int t = threadIdx.x;
  const float* tile = uhat + (size_t)c * TILE_F + n * 2048;
  const float* vt = vT + n * 2048;
  float p = 0.0f;
  #pragma unroll
  for (int j = 0; j < 8; ++j) {
    const int e = t + 256 * j;
    p = fmaf(tile[e], vt[e], p);
  }
  __shared__ float red[256];
  red[t] = p;
  __syncthreads();
  for (int s = 128; s > 0; s >>= 1) {
    if (t < s) red[t] += red[t + s];
    __syncthreads();
  }
  if (t == 0) bij[c * N_CAP + n] += red[0] * (1.0f / (float)B_SZ);
}

// ---------------------------------------------------------------------------
extern "C" void kernel_launch(void* const* d_in, const int* in_sizes, int n_in,
                              void* d_out, int out_size, void* d_ws, size_t ws_size,
                              hipStream_t stream) {
  const float* x = (const float*)d_in[0];   // [32,64,1024]
  const float* W = (const float*)d_in[1];   // [1,1024,32,64,64]
  float* out = (float*)d_out;               // [32,32,64,1]

  float* ws = (float*)d_ws;
  float* uhat    = ws;                                   // 67,108,864 floats
  float* bij     = uhat + (size_t)C_CH * TILE_F;         // 32,768
  float* cij     = bij + C_CH * N_CAP;                   // 32,768
  float* partial = cij + C_CH * N_CAP;                   // 2,097,152
  float* sT      = partial + 32 * N_CAP * 2048;          // 65,536
  float* vT      = sT + N_CAP * 2048;                    // 65,536

  // Fresh b_ij every launch (ws is not re-poisoned between replays).
  zero_bij_kernel<<<(C_CH * N_CAP + 255) / 256, 256, 0, stream>>>(bij);

  // u_hat GEMM: one block per channel.
  uhat_wmma_kernel<<<C_CH, 256, 0, stream>>>(x, W, uhat);

  for (int it = 0; it < NUM_IT; ++it) {
    softmax_kernel<<<(C_CH + 255) / 256, 256, 0, stream>>>(bij, cij);
    s_partial_kernel<<<N_CAP * 32, 256, 0, stream>>>(uhat, cij, partial);
    s_reduce_kernel<<<(N_CAP * 2048) / 256, 256, 0, stream>>>(partial, sT);
    squash_kernel<<<(2048) / 256, 256, 0, stream>>>(sT, vT, out);
    if (it < NUM_IT - 1) {
      // last iteration's agreement never affects the output -> skip 256 MB read
      agree_kernel<<<C_CH * N_CAP, 256, 0, stream>>>(uhat, vT, bij);
    }
  }
}